// NTM_25778393711122
// MI455X (gfx1250) — compile-verified
//
#include <hip/hip_runtime.h>
#include <cstddef>

// ---------------------------------------------------------------------------
// NTM single step for MI455X (gfx1250), f32 throughout.
// Pure bandwidth problem (~300MB weight streaming @ 23.3 TB/s ~= 13us).
// GEMVs use V_WMMA_F32_16X16X4_F32 with a broadcast-A trick (16 rows/wave),
// b128 per-lane loads via K-slot permutation, and NT loads for the
// single-use 256MB we_w stream (preserves reusable weights in 192MB L2).
// ---------------------------------------------------------------------------

typedef __attribute__((ext_vector_type(2))) float v2f;
typedef __attribute__((ext_vector_type(4))) float v4f;
typedef __attribute__((ext_vector_type(8))) float v8f;

#define Hdim 1024
#define Ndim 128
#define Mdim 512

// d_out layout (floats): out | rw | ww | h | c | new_mem
#define OUT_OFF 0
#define RW_OFF  1024
#define WW_OFF  1152
#define H_OFF   1280
#define C_OFF   2304
#define MEM_OFF 3328

// workspace layout (floats)
#define WS_GATES 0        // 4096
#define WS_H     4096     // 1024
#define WS_KR    5120     // 512
#define WS_KW    5632     // 512
#define WS_SR    6144     // 128
#define WS_SW    6272     // 128
#define WS_SCAL  6400     // 8  ([0..2] read beta/g/gamma pre, [3..5] write)
#define WS_READ  6408     // 512
#define WS_A     6920     // 512
#define WS_ERASE 7432     // 512
#define WS_FCIN  7944     // 1536
#define WS_E     9480     // 65536

__device__ __forceinline__ v8f wmma_f32_4(v2f a, v2f b, v8f c) {
  // D = A(16x4, broadcast rows) x B(4x16) + C
  // (neg_a, A, neg_b, B, c_mod, C, reuse_a, reuse_b)
  return __builtin_amdgcn_wmma_f32_16x16x4_f32(false, a, false, b,
                                               (short)0, c, false, false);
}

template <bool NT>
__device__ __forceinline__ v4f load4(const float* p) {
  if (NT) return __builtin_nontemporal_load((const v4f*)p);
  return *(const v4f*)p;
}

__device__ __forceinline__ float sigmoidf_(float z) {
  return 1.0f / (1.0f + expf(-z));
}
__device__ __forceinline__ float softplusf_(float z) {
  return (z > 20.0f) ? z : log1pf(expf(z));
}

// y[r] = dot(W[r, 0:cols], x) + bias[r]  (+optional sigmoid). cols % 8 == 0.
// One wave per 16-row tile. K-slot permutation: per macro-step of 8 K values,
// lanes 0-15 feed K {k0..k0+3}, lanes 16-31 feed K {k0+4..k0+7}, identically
// for A and B, so the accumulated dot product is exact (reordered only).
template <bool NT>
__global__ void gemv_wmma_kernel(const float* __restrict__ W,
                                 const float* __restrict__ bias,
                                 const float* __restrict__ x,
                                 float* __restrict__ y,
                                 int rows, int cols, int act) {
  const int lane = threadIdx.x;                  // 0..31
  const int r    = blockIdx.x * 16 + (lane & 15);
  const int rc   = (r < rows) ? r : (rows - 1);  // clamp for guarded tiles
  const float* wrow = W + (size_t)rc * (size_t)cols;
  const int boff = (lane >> 4) * 4;              // lanes 0-15: +0, 16-31: +4
  v8f acc = {};
  for (int k0 = 0; k0 < cols; k0 += 8) {
    v4f a4 = *(const v4f*)(x + k0 + boff);
    v4f b4 = load4<NT>(wrow + k0 + boff);
    v2f a1 = {a4.x, a4.y};
    v2f b1 = {b4.x, b4.y};
    v2f a2 = {a4.z, a4.w};
    v2f b2 = {b4.z, b4.w};
    acc = wmma_f32_4(a1, b1, acc);
    acc = wmma_f32_4(a2, b2, acc);
  }
  if (lane < 16 && r < rows) {
    float v = acc[0] + bias[r];                  // D[0, lane]
    if (act == 1) v = sigmoidf_(v);
    y[r] = v;
  }
}

// gates[r] = dot(W_ih[r],x) + b_ih[r] + dot(W_hh[r],h0) + b_hh[r], rows=4096
__global__ void lstm_gates_kernel(const float* __restrict__ Wih,
                                  const float* __restrict__ Whh,
                                  const float* __restrict__ bih,
                                  const float* __restrict__ bhh,
                                  const float* __restrict__ x,
                                  const float* __restrict__ h0,
                                  float* __restrict__ gates) {
  const int lane = threadIdx.x;
  const int r    = blockIdx.x * 16 + (lane & 15);
  const float* wi = Wih + (size_t)r * Hdim;
  const float* wh = Whh + (size_t)r * Hdim;
  const int boff = (lane >> 4) * 4;
  v8f acc1 = {};
  v8f acc2 = {};
  for (int k0 = 0; k0 < Hdim; k0 += 8) {
    v4f ax = *(const v4f*)(x  + k0 + boff);
    v4f ah = *(const v4f*)(h0 + k0 + boff);
    v4f bi = *(const v4f*)(wi + k0 + boff);
    v4f bh = *(const v4f*)(wh + k0 + boff);
    v2f a1 = {ax.x, ax.y}; v2f a2 = {ax.z, ax.w};
    v2f h1 = {ah.x, ah.y}; v2f h2 = {ah.z, ah.w};
    v2f i1 = {bi.x, bi.y}; v2f i2 = {bi.z, bi.w};
    v2f r1 = {bh.x, bh.y}; v2f r2 = {bh.z, bh.w};
    acc1 = wmma_f32_4(a1, i1, acc1);
    acc1 = wmma_f32_4(a2, i2, acc1);
    acc2 = wmma_f32_4(h1, r1, acc2);
    acc2 = wmma_f32_4(h2, r2, acc2);
  }
  if (lane < 16) gates[r] = acc1[0] + acc2[0] + bih[r] + bhh[r];
}

// torch gate order i,f,g,o
__global__ void lstm_cell_kernel(const float* __restrict__ gates,
                                 const float* __restrict__ c0,
                                 float* __restrict__ h_ws,
                                 float* __restrict__ h_out,
                                 float* __restrict__ c_out) {
  const int j = blockIdx.x * blockDim.x + threadIdx.x;
  if (j >= Hdim) return;
  const float gi = gates[j];
  const float gf = gates[Hdim + j];
  const float gg = gates[2 * Hdim + j];
  const float go = gates[3 * Hdim + j];
  const float c  = sigmoidf_(gf) * c0[j] + sigmoidf_(gi) * tanhf(gg);
  const float h  = sigmoidf_(go) * tanhf(c);
  h_ws[j]  = h;
  h_out[j] = h;
  c_out[j] = c;
}

__device__ __forceinline__ float blk_reduce_sum(float v, float* red, int tid) {
  __syncthreads();
  red[tid] = v;
  __syncthreads();
  for (int s = Ndim >> 1; s > 0; s >>= 1) {
    if (tid < s) red[tid] += red[tid + s];
    __syncthreads();
  }
  float r = red[0];
  __syncthreads();
  return r;
}

__device__ __forceinline__ float blk_reduce_max(float v, float* red, int tid) {
  __syncthreads();
  red[tid] = v;
  __syncthreads();
  for (int s = Ndim >> 1; s > 0; s >>= 1) {
    if (tid < s) red[tid] = fmaxf(red[tid], red[tid + s]);
    __syncthreads();
  }
  float r = red[0];
  __syncthreads();
  return r;
}

// One block of 128 threads: full NTM addressing for one head.
// scal: [0]=beta_pre [1]=g_pre [2]=gamma_pre
__global__ void address_kernel(const float* __restrict__ kvec,
                               const float* __restrict__ scal,
                               const float* __restrict__ s_logits,
                               const float* __restrict__ mem,
                               const float* __restrict__ prev_w,
                               float* __restrict__ w_out) {
  __shared__ float red[Ndim];
  __shared__ float wg[Ndim];
  __shared__ float ss[Ndim];
  __shared__ float sc[4];
  const int n = threadIdx.x;  // 0..127

  // ||k|| via 128x4 partials
  float kp = 0.0f;
  for (int t = 0; t < 4; ++t) {
    const float kv = kvec[n * 4 + t];
    kp += kv * kv;
  }
  const float ksum = blk_reduce_sum(kp, red, n);
  if (n == 0) {
    sc[0] = softplusf_(scal[0]);          // beta
    sc[1] = sigmoidf_(scal[1]);           // g
    sc[2] = 1.0f + softplusf_(scal[2]);   // gamma
    sc[3] = fmaxf(sqrtf(ksum), 1e-8f);    // ||k||
  }
  __syncthreads();

  // cosine similarity of memory row n with k
  float dot = 0.0f, nrm = 0.0f;
  const float* row = mem + (size_t)n * Mdim;
  for (int m = 0; m < Mdim; ++m) {
    const float mv = row[m];
    dot += mv * kvec[m];
    nrm += mv * mv;
  }
  const float cosv = dot / (fmaxf(sqrtf(nrm), 1e-8f) * sc[3]);

  // w_c = softmax(beta * cos)
  const float z    = sc[0] * cosv;
  const float zmax = blk_reduce_max(z, red, n);
  const float ez   = expf(z - zmax);
  const float esum = blk_reduce_sum(ez, red, n);
  const float wc   = ez / esum;

  // interpolation (write head also receives prev_read_w, faithful to source)
  const float g = sc[1];
  wg[n] = g * wc + (1.0f - g) * prev_w[n];

  // s = softmax(s_logits)
  const float sl   = s_logits[n];
  const float smax = blk_reduce_max(sl, red, n);
  const float es   = expf(sl - smax);
  const float ssum = blk_reduce_sum(es, red, n);
  ss[n] = es / ssum;
  __syncthreads();

  // circular conv: w_t[n] = sum_j wg[j] * s[(n - j) mod N]
  float wt = 0.0f;
  for (int j = 0; j < Ndim; ++j) wt += wg[j] * ss[(n - j) & (Ndim - 1)];

  // sharpen + renormalize
  const float w    = powf(wt, sc[2]);
  const float wsum = blk_reduce_sum(w, red, n);
  w_out[n] = w / (wsum + 1e-16f);
}

// out[m] = sum_n w[n] * mat[n, m]   (column-weighted sum; coalesced over m)
__global__ void weighted_rows_kernel(const float* __restrict__ w,
                                     const float* __restrict__ mat,
                                     float* __restrict__ out, int N, int M) {
  const int m = blockIdx.x * blockDim.x + threadIdx.x;
  if (m >= M) return;
  float acc = 0.0f;
  for (int n = 0; n < N; ++n) acc += w[n] * mat[(size_t)n * M + m];
  out[m] = acc;
}

// new_mem[n,m] = memory[n,m]*(1 - erase[m]) + ww[n]*a[m]
// (erase is the (1,M) row vector ww @ e.reshape(N,M), faithful to reference)
__global__ void mem_update_kernel(const float* __restrict__ memory,
                                  const float* __restrict__ erase,
                                  const float* __restrict__ ww,
                                  const float* __restrict__ a,
                                  float* __restrict__ out) {
  const int idx = blockIdx.x * blockDim.x + threadIdx.x;
  if (idx >= Ndim * Mdim) return;
  const int n = idx >> 9;       // / 512
  const int m = idx & (Mdim - 1);
  out[idx] = memory[idx] * (1.0f - erase[m]) + ww[n] * a[m];
}

__global__ void concat_kernel(const float* __restrict__ h,
                              const float* __restrict__ rd,
                              float* __restrict__ fc_in) {
  const int i = blockIdx.x * blockDim.x + threadIdx.x;
  if (i < Hdim) fc_in[i] = h[i];
  else if (i < Hdim + Mdim) fc_in[i] = rd[i - Hdim];
}

extern "C" void kernel_launch(void* const* d_in, const int* in_sizes, int n_in,
                              void* d_out, int out_size, void* d_ws, size_t ws_size,
                              hipStream_t stream) {
  const float* x           = (const float*)d_in[0];
  const float* prev_read_w = (const float*)d_in[1];
  // d_in[2] = prev_write_w (unused: reference passes prev_read_w to BOTH heads)
  const float* h0      = (const float*)d_in[3];
  const float* c0      = (const float*)d_in[4];
  const float* memory  = (const float*)d_in[5];
  const float* W_ih    = (const float*)d_in[6];
  const float* W_hh    = (const float*)d_in[7];
  const float* b_ih    = (const float*)d_in[8];
  const float* b_hh    = (const float*)d_in[9];
  const float* rk_w    = (const float*)d_in[10];
  const float* rk_b    = (const float*)d_in[11];
  const float* rbeta_w = (const float*)d_in[12];
  const float* rbeta_b = (const float*)d_in[13];
  const float* rg_w    = (const float*)d_in[14];
  const float* rg_b    = (const float*)d_in[15];
  const float* rs_w    = (const float*)d_in[16];
  const float* rs_b    = (const float*)d_in[17];
  const float* rgamma_w= (const float*)d_in[18];
  const float* rgamma_b= (const float*)d_in[19];
  const float* wk_w    = (const float*)d_in[20];
  const float* wk_b    = (const float*)d_in[21];
  const float* wbeta_w = (const float*)d_in[22];
  const float* wbeta_b = (const float*)d_in[23];
  const float* wg_w    = (const float*)d_in[24];
  const float* wg_b    = (const float*)d_in[25];
  const float* ws_w    = (const float*)d_in[26];
  const float* ws_b    = (const float*)d_in[27];
  const float* wgamma_w= (const float*)d_in[28];
  const float* wgamma_b= (const float*)d_in[29];
  const float* we_w    = (const float*)d_in[30];
  const float* we_b    = (const float*)d_in[31];
  const float* wa_w    = (const float*)d_in[32];
  const float* wa_b    = (const float*)d_in[33];
  const float* fc_w    = (const float*)d_in[34];
  const float* fc_b    = (const float*)d_in[35];

  float* out = (float*)d_out;
  float* ws  = (float*)d_ws;
  float* gates = ws + WS_GATES;
  float* hbuf  = ws + WS_H;
  float* kr    = ws + WS_KR;
  float* kw    = ws + WS_KW;
  float* sr    = ws + WS_SR;
  float* sw    = ws + WS_SW;
  float* scal  = ws + WS_SCAL;
  float* rdbuf = ws + WS_READ;
  float* abuf  = ws + WS_A;
  float* ebuf  = ws + WS_E;
  float* erase = ws + WS_ERASE;
  float* fcin  = ws + WS_FCIN;

  // 1) LSTM controller
  lstm_gates_kernel<<<4 * Hdim / 16, 32, 0, stream>>>(W_ih, W_hh, b_ih, b_hh,
                                                      x, h0, gates);
  lstm_cell_kernel<<<(Hdim + 255) / 256, 256, 0, stream>>>(gates, c0, hbuf,
                                                           out + H_OFF, out + C_OFF);

  // 2) read-head parameter GEMVs (pre-activations; addressing applies acts)
  gemv_wmma_kernel<false><<<Mdim / 16, 32, 0, stream>>>(rk_w, rk_b, hbuf, kr, Mdim, Hdim, 0);
  gemv_wmma_kernel<false><<<1, 32, 0, stream>>>(rbeta_w,  rbeta_b,  hbuf, scal + 0, 1, Hdim, 0);
  gemv_wmma_kernel<false><<<1, 32, 0, stream>>>(rg_w,     rg_b,     hbuf, scal + 1, 1, Hdim, 0);
  gemv_wmma_kernel<false><<<Ndim / 16, 32, 0, stream>>>(rs_w, rs_b, hbuf, sr, Ndim, Hdim, 0);
  gemv_wmma_kernel<false><<<1, 32, 0, stream>>>(rgamma_w, rgamma_b, hbuf, scal + 2, 1, Hdim, 0);
  address_kernel<<<1, Ndim, 0, stream>>>(kr, scal, sr, memory, prev_read_w, out + RW_OFF);

  // 3) write-head parameter GEMVs + addressing (also fed prev_read_w)
  gemv_wmma_kernel<false><<<Mdim / 16, 32, 0, stream>>>(wk_w, wk_b, hbuf, kw, Mdim, Hdim, 0);
  gemv_wmma_kernel<false><<<1, 32, 0, stream>>>(wbeta_w,  wbeta_b,  hbuf, scal + 3, 1, Hdim, 0);
  gemv_wmma_kernel<false><<<1, 32, 0, stream>>>(wg_w,     wg_b,     hbuf, scal + 4, 1, Hdim, 0);
  gemv_wmma_kernel<false><<<Ndim / 16, 32, 0, stream>>>(ws_w, ws_b, hbuf, sw, Ndim, Hdim, 0);
  gemv_wmma_kernel<false><<<1, 32, 0, stream>>>(wgamma_w, wgamma_b, hbuf, scal + 5, 1, Hdim, 0);
  address_kernel<<<1, Ndim, 0, stream>>>(kw, scal + 3, sw, memory, prev_read_w, out + WW_OFF);

  // 4) read vector: read = rw @ mem
  weighted_rows_kernel<<<(Mdim + 255) / 256, 256, 0, stream>>>(out + RW_OFF, memory,
                                                               rdbuf, Ndim, Mdim);

  // 5) erase / add vectors (the 256 MB single-use stream: NT loads)
  gemv_wmma_kernel<true><<<(Ndim * Mdim) / 16, 32, 0, stream>>>(we_w, we_b, hbuf, ebuf,
                                                                Ndim * Mdim, Hdim, 1);
  gemv_wmma_kernel<false><<<Mdim / 16, 32, 0, stream>>>(wa_w, wa_b, hbuf, abuf, Mdim, Hdim, 0);

  // 6) memory update: erase[m] = sum_n ww[n]*e[n,m]; then elementwise update
  weighted_rows_kernel<<<(Mdim + 255) / 256, 256, 0, stream>>>(out + WW_OFF, ebuf,
                                                               erase, Ndim, Mdim);
  mem_update_kernel<<<(Ndim * Mdim + 255) / 256, 256, 0, stream>>>(memory, erase,
                                                                   out + WW_OFF, abuf,
                                                                   out + MEM_OFF);

  // 7) output projection: out = fc_w @ [h ; read] + fc_b
  concat_kernel<<<(Hdim + Mdim + 255) / 256, 256, 0, stream>>>(hbuf, rdbuf, fcin);
  gemv_wmma_kernel<false><<<Hdim / 16, 32, 0, stream>>>(fc_w, fc_b, fcin, out + OUT_OFF,
                                                        Hdim, Hdim + Mdim, 0);

  (void)in_sizes; (void)n_in; (void)out_size; (void)ws_size;
}